// DCRNN_model_78039555769038
// MI455X (gfx1250) — compile-verified
//
#include <hip/hip_runtime.h>
#include <hip/hip_bf16.h>

typedef __attribute__((ext_vector_type(2))) float v2f;
typedef __attribute__((ext_vector_type(8))) float v8f;

// ---------------------------------------------------------------------------
// Prep: fold Wz/Wh (2,288,32) -> effective [256,64] (cols 0-31 = Wz_eff,
// cols 32-63 = Wh_eff) and pre-swizzle into B-fragment layout for
// V_WMMA_F32_16X16X4_F32:
//   bfrag[s][t][lane][0..1]  (s = k-step 0..63, t = 16-col tile 0..3)
//   lane<16 : {W[4s+0][c], W[4s+1][c]}   c = t*16 + lane
//   lane>=16: {W[4s+2][c], W[4s+3][c]}   c = t*16 + lane-16
// Total 64*4*32*2 floats = 64 KB in d_ws.
// ---------------------------------------------------------------------------
__global__ void dcrnn_prep_bfrag(const float* __restrict__ Wz,
                                 const float* __restrict__ Wh,
                                 float* __restrict__ bfrag) {
    int idx = blockIdx.x * blockDim.x + threadIdx.x;   // 0 .. 8191
    if (idx >= 64 * 4 * 32) return;
    int lane = idx & 31;
    int t    = (idx >> 5) & 3;
    int s    = idx >> 7;
    int col  = t * 16 + (lane & 15);                   // 0..63
    int k0   = s * 4 + (lane >> 4) * 2;                // base K of this half
    const float* W = (col < 32) ? Wz : Wh;             // (2,288,32) row-major
    int j = col & 31;
    // eff[k][j] = W[0][k][j] + W[1][k][j], k < 256
    float v0 = W[(size_t)k0 * 32 + j]       + W[288u * 32 + (size_t)k0 * 32 + j];
    float v1 = W[(size_t)(k0 + 1) * 32 + j] + W[288u * 32 + (size_t)(k0 + 1) * 32 + j];
    bfrag[(size_t)idx * 2 + 0] = v0;
    bfrag[(size_t)idx * 2 + 1] = v1;
}

__device__ __forceinline__ float sigmoidf_(float v) {
    return 1.0f / (1.0f + __expf(-v));
}

// ---------------------------------------------------------------------------
// Main: 8 waves / block, each wave computes a 16-row stripe.
//   acc0/acc1 : x @ Wz_eff   cols 0-15 / 16-31
//   acc2/acc3 : x @ Wh_eff   cols 0-15 / 16-31
// then fused sigmoid/tanh/relu + dot with Wl + butterfly reduce + store.
// ---------------------------------------------------------------------------
__global__ __launch_bounds__(256) void dcrnn_main(
    const float* __restrict__ x,      // [N,256]
    const float* __restrict__ bfrag,  // [64][4][32][2]
    const float* __restrict__ bz,     // [32]
    const float* __restrict__ bh,     // [32]
    const float* __restrict__ Wl,     // [32]
    const float* __restrict__ bl,     // [1]
    float* __restrict__ out,          // [N]
    int nRows)
{
    const int lane = threadIdx.x & 31;
    const int wave = threadIdx.x >> 5;
    const int m    = lane & 15;       // col within tile / row within stripe
    const int half = lane >> 4;       // 0: K{0,1} rows 0-7 ; 1: K{2,3} rows 8-15
    const int rowBase = (blockIdx.x * 8 + wave) * 16;

    // Clamp A-load row so EXEC can stay all-1s (WMMA requirement); stores guard.
    int arow = rowBase + m;
    if (arow >= nRows) arow = nRows - 1;
    const float* xrow  = x + (size_t)arow * 256 + half * 2;
    const float* bbase = bfrag + (size_t)lane * 2;

    v8f acc0 = {}, acc1 = {}, acc2 = {}, acc3 = {};

#pragma unroll 4
    for (int s = 0; s < 64; ++s) {
        // A fragment: lane<16 -> x[row][4s+0,4s+1], lane>=16 -> x[row][4s+2,4s+3]
        v2f a = *(const v2f*)(xrow + s * 4);
        const float* bs = bbase + (size_t)s * 256;     // 4 tiles * 64 floats
        v2f b0 = *(const v2f*)(bs + 0 * 64);
        v2f b1 = *(const v2f*)(bs + 1 * 64);
        v2f b2 = *(const v2f*)(bs + 2 * 64);
        v2f b3 = *(const v2f*)(bs + 3 * 64);
        acc0 = __builtin_amdgcn_wmma_f32_16x16x4_f32(false, a, false, b0, (short)0, acc0, false, false);
        acc1 = __builtin_amdgcn_wmma_f32_16x16x4_f32(false, a, false, b1, (short)0, acc1, false, false);
        acc2 = __builtin_amdgcn_wmma_f32_16x16x4_f32(false, a, false, b2, (short)0, acc2, false, false);
        acc3 = __builtin_amdgcn_wmma_f32_16x16x4_f32(false, a, false, b3, (short)0, acc3, false, false);
    }

    // Per-lane columns: tile0/2 -> col m, tile1/3 -> col 16+m
    const float bz0 = bz[m],      bz1 = bz[16 + m];
    const float bh0 = bh[m],      bh1 = bh[16 + m];
    const float wl0 = Wl[m],      wl1 = Wl[16 + m];

    float partial[8];
#pragma unroll
    for (int r = 0; r < 8; ++r) {
        float z0 = sigmoidf_(acc0[r] + bz0);
        float z1 = sigmoidf_(acc1[r] + bz1);
        float t0 = tanhf(acc2[r] + bh0);
        float t1 = tanhf(acc3[r] + bh1);
        float h0 = fmaxf(0.0f, (1.0f - z0) * t0);
        float h1 = fmaxf(0.0f, (1.0f - z1) * t1);
        partial[r] = h0 * wl0 + h1 * wl1;
    }

    // Reduce over the 16 lanes of each half (rows 0-7 live in lanes 0-15,
    // rows 8-15 in lanes 16-31 per the C/D layout).
#pragma unroll
    for (int mask = 1; mask < 16; mask <<= 1) {
#pragma unroll
        for (int r = 0; r < 8; ++r)
            partial[r] += __shfl_xor(partial[r], mask, 32);
    }

    if (m == 0) {
        const float blv = bl[0];
        const int r0 = rowBase + half * 8;
#pragma unroll
        for (int r = 0; r < 8; ++r) {
            int row = r0 + r;
            if (row < nRows) out[row] = partial[r] + blv;
        }
    }
}

extern "C" void kernel_launch(void* const* d_in, const int* in_sizes, int n_in,
                              void* d_out, int out_size, void* d_ws, size_t ws_size,
                              hipStream_t stream) {
    // setup_inputs order: x, edge_index(unused), Wz, bz, Wr(unused), br(unused),
    //                     Wh, bh, Wl, bl
    const float* x   = (const float*)d_in[0];
    const float* Wz  = (const float*)d_in[2];
    const float* bz  = (const float*)d_in[3];
    const float* Wh  = (const float*)d_in[6];
    const float* bh  = (const float*)d_in[7];
    const float* Wl  = (const float*)d_in[8];
    const float* bl  = (const float*)d_in[9];
    float* out   = (float*)d_out;
    float* bfrag = (float*)d_ws;                 // 64 KB used

    const int nRows = in_sizes[0] / 256;         // N = 200000

    dcrnn_prep_bfrag<<<32, 256, 0, stream>>>(Wz, Wh, bfrag);

    const int rowsPerBlock = 128;                // 8 waves * 16 rows
    const int nb = (nRows + rowsPerBlock - 1) / rowsPerBlock;
    dcrnn_main<<<nb, 256, 0, stream>>>(x, bfrag, bz, bh, Wl, bl, out, nRows);
}